// OccupancyManager_56573309224608
// MI455X (gfx1250) — compile-verified
//
#include <hip/hip_runtime.h>

#define BLOCK   256
#define NLEV    16
#define TSIZE   524288u
#define TMASK   524287u
#define ROWP    34           // padded LDS row stride (floats): stride-34 -> conflict-free,
                             // and 34*lane + 2l is even -> float2 writes are 8B aligned

typedef __attribute__((ext_vector_type(4))) float v4f;
typedef __attribute__((ext_vector_type(2))) float v2f;

__global__ __launch_bounds__(BLOCK) void hashgrid_fwd(
    const float* __restrict__ xyz,
    const float* __restrict__ tables,
    const int*   __restrict__ resolutions,
    float*       __restrict__ out, int n)
{
    __shared__ float s_xyz[BLOCK * 3];              // 3 KB
    __shared__ float s_out[8 * 32 * ROWP];          // 34 KB: per-wave 32x32 tile (+pad)

    const int tid  = threadIdx.x;
    const int wave = tid >> 5;
    const int lane = tid & 31;
    const long long blockBase = (long long)blockIdx.x * BLOCK;

    // ---- Stage xyz tile into LDS via CDNA5 async global->LDS DMA path ----
    {
        const float* gsrc = xyz + blockBase * 3;
        const long long limit = (long long)n * 3 - blockBase * 3;
        #pragma unroll
        for (int i = 0; i < 3; ++i) {
            int k = i * BLOCK + tid;
            if (k < limit) {
                unsigned lds_off = (unsigned)(size_t)(&s_xyz[k]);
                unsigned long long ga = (unsigned long long)(size_t)(gsrc + k);
                asm volatile("global_load_async_to_lds_b32 %0, %1, off"
                             :: "v"(lds_off), "v"(ga) : "memory");
            }
        }
    }
    asm volatile("s_wait_asynccnt 0" ::: "memory");
    __syncthreads();

    float* so = &s_out[wave * (32 * ROWP)];
    const long long gid = blockBase + tid;

    if (gid < n) {
        const float px = fminf(fmaxf(s_xyz[tid * 3 + 0] + 0.5f, 0.0f), 1.0f - 1e-6f);
        const float py = fminf(fmaxf(s_xyz[tid * 3 + 1] + 0.5f, 0.0f), 1.0f - 1e-6f);
        const float pz = fminf(fmaxf(s_xyz[tid * 3 + 2] + 0.5f, 0.0f), 1.0f - 1e-6f);

        // unroll 2: keep two levels' gather batches in flight per wave (MLP for a
        // latency-bound kernel) without blowing the VGPR budget / occupancy.
        #pragma unroll 2
        for (int l = 0; l < NLEV; ++l) {
            const float rf = (float)resolutions[l];
            const float2* __restrict__ tab = (const float2*)tables + (size_t)l * TSIZE;

            const float fx = px * rf, fy = py * rf, fz = pz * rf;
            const float xf = floorf(fx), yf = floorf(fy), zf = floorf(fz);
            const float wx = fx - xf, wy = fy - yf, wz = fz - zf;
            const unsigned ux = (unsigned)xf, uy = (unsigned)yf, uz = (unsigned)zf;

            // hash: x*1 ^ y*2654435761 ^ z*805459861, mod 2^19 == AND
            const unsigned hy0 = uy * 2654435761u;
            const unsigned hy1 = hy0 + 2654435761u;      // (uy+1)*P1 mod 2^32
            const unsigned hz0 = uz * 805459861u;
            const unsigned hz1 = hz0 + 805459861u;       // (uz+1)*P2 mod 2^32
            const unsigned x0 = ux, x1 = ux + 1u;

            const unsigned e00 = x0 ^ hy0, e01 = x0 ^ hy1;
            const unsigned e10 = x1 ^ hy0, e11 = x1 ^ hy1;

            const unsigned i000 = (e00 ^ hz0) & TMASK;
            const unsigned i001 = (e00 ^ hz1) & TMASK;
            const unsigned i010 = (e01 ^ hz0) & TMASK;
            const unsigned i011 = (e01 ^ hz1) & TMASK;
            const unsigned i100 = (e10 ^ hz0) & TMASK;
            const unsigned i101 = (e10 ^ hz1) & TMASK;
            const unsigned i110 = (e11 ^ hz0) & TMASK;
            const unsigned i111 = (e11 ^ hz1) & TMASK;

            // 8 independent b64 gathers (tables are L2-resident: 64MB of 192MB)
            const float2 f000 = tab[i000];
            const float2 f001 = tab[i001];
            const float2 f010 = tab[i010];
            const float2 f011 = tab[i011];
            const float2 f100 = tab[i100];
            const float2 f101 = tab[i101];
            const float2 f110 = tab[i110];
            const float2 f111 = tab[i111];

            const float ax = 1.0f - wx, ay = 1.0f - wy, az = 1.0f - wz;
            const float w00 = ax * ay, w01 = ax * wy, w10 = wx * ay, w11 = wx * wy;
            const float w000 = w00 * az, w001 = w00 * wz;
            const float w010 = w01 * az, w011 = w01 * wz;
            const float w100 = w10 * az, w101 = w10 * wz;
            const float w110 = w11 * az, w111 = w11 * wz;

            const float r0 = w000 * f000.x + w001 * f001.x + w010 * f010.x + w011 * f011.x
                           + w100 * f100.x + w101 * f101.x + w110 * f110.x + w111 * f111.x;
            const float r1 = w000 * f000.y + w001 * f001.y + w010 * f010.y + w011 * f011.y
                           + w100 * f100.y + w101 * f101.y + w110 * f110.y + w111 * f111.y;

            // 8B-aligned, bank-conflict-free ds_store_b64
            v2f r = { r0, r1 };
            *(v2f*)(&so[lane * ROWP + 2 * l]) = r;
        }
    }
    __syncthreads();

    // ---- Coalesced write-out: each wave streams its 32x32 tile as b128 NT stores ----
    const long long rowbase = blockBase + (long long)wave * 32;
    #pragma unroll
    for (int j = 0; j < 8; ++j) {
        const int row = j * 4 + (lane >> 3);
        const int col = (lane & 7) * 4;
        if (rowbase + row < n) {
            const float* src = &so[row * ROWP + col];
            v4f v = { src[0], src[1], src[2], src[3] };
            __builtin_nontemporal_store(v, (v4f*)(out + (size_t)(rowbase + row) * 32 + col));
        }
    }
}

extern "C" void kernel_launch(void* const* d_in, const int* in_sizes, int n_in,
                              void* d_out, int out_size, void* d_ws, size_t ws_size,
                              hipStream_t stream) {
    const float* xyz         = (const float*)d_in[0];   // [N,3] f32
    const float* tables      = (const float*)d_in[1];   // [16, 524288, 2] f32
    const int*   resolutions = (const int*)d_in[2];     // [16] i32
    float*       out         = (float*)d_out;           // [N, 32] f32

    const int n = in_sizes[0] / 3;
    const int blocks = (n + BLOCK - 1) / BLOCK;
    hashgrid_fwd<<<blocks, BLOCK, 0, stream>>>(xyz, tables, resolutions, out, n);
}